// DifferentialMultiHeadAttention_37692632990208
// MI455X (gfx1250) — compile-verified
//
#include <hip/hip_runtime.h>
#include <cmath>

// ---------------------------------------------------------------------------
// DifferentialMultiHeadAttention forward for MI455X (gfx1250, wave32).
// All matmuls run on V_WMMA_F32_16X16X4_F32 (native fp32 matrix pipe) to
// match the fp32 reference exactly up to FMA-ordering. Tile staging uses the
// CDNA5 async global->LDS DMA path (ASYNCcnt) when the toolchain exposes it.
// ---------------------------------------------------------------------------

typedef __attribute__((ext_vector_type(2))) float v2f;
typedef __attribute__((ext_vector_type(4))) float v4f;
typedef __attribute__((ext_vector_type(8))) float v8f;
typedef int v4i __attribute__((vector_size(16)));   // matches builtin proto

#define DIMD 1024
#define NH   16
#define HDM  64
#define BB   2
#define SS   2048
#define MROWS (BB * SS)      // 4096
#define HSTR  (NH * HDM)     // 1024 row stride of [b,s,h,hd]

#if __has_builtin(__builtin_amdgcn_global_load_async_to_lds_b128) && \
    __has_builtin(__builtin_amdgcn_s_wait_asynccnt)
#define HAVE_ASYNC_LDS 1
#else
#define HAVE_ASYNC_LDS 0
#endif

// Builtin proto (from hipcc diagnostic): (v4i AS1*, v4i AS3*, Ii offset, Ii cpol)
#define AS_GLOBAL(p) ((__attribute__((address_space(1))) v4i*)(p))
#define AS_LDS(p)    ((__attribute__((address_space(3))) v4i*)(p))

static __device__ __forceinline__ v8f wmma4(v2f a, v2f b, v8f c) {
  // D = A(16x4,f32) * B(4x16,f32) + C(16x16,f32)
  return __builtin_amdgcn_wmma_f32_16x16x4_f32(
      /*neg_a=*/false, a, /*neg_b=*/false, b,
      /*c_mod=*/(short)0, c, /*reuse_a=*/false, /*reuse_b=*/false);
}

// ---------------------------------------------------------------------------
// C[M,N] = A[M,K] @ B[K,N] + bias[N]      (fp32, WMMA 16x16x4)
// Block: 256 threads = 8 waves; 64x64 C tile; each wave owns a 16x32 strip.
// ---------------------------------------------------------------------------
__global__ __launch_bounds__(256) void gemm_bias_kernel(
    const float* __restrict__ A, const float* __restrict__ Bm,
    const float* __restrict__ bias, float* __restrict__ C,
    int M, int N, int K)
{
  // Padded strides: sA row = 20 floats (80B: 16B-aligned b128 DMA chunks,
  // aligned b64 frag reads, bank spread), sB row = 80 floats (320B).
  __shared__ float sA[64 * 20];
  __shared__ float sB[16 * 80];

  const int tid  = threadIdx.x;
  const int lane = tid & 31;
  const int wid  = tid >> 5;
  const int lm   = lane & 15;      // M (A/C) or N (B/C) index within frag
  const int lh   = lane >> 4;      // half-wave select
  const int wm   = (wid & 3) * 16; // wave row offset inside 64x64 tile
  const int wn   = (wid >> 2) * 32;// wave col offset
  const int bm   = blockIdx.y * 64;
  const int bn   = blockIdx.x * 64;

  v8f acc0 = {};
  v8f acc1 = {};

  // cooperative tile-load mapping (one 16B chunk per thread per panel)
  const int ar = tid >> 2;          // 0..63
  const int ac = (tid & 3) * 4;     // 0,4,8,12
  const int br = tid >> 4;          // 0..15
  const int bc = (tid & 15) * 4;    // 0..60

  for (int kb = 0; kb < K; kb += 16) {
    if (kb + 16 < K) {
      // CDNA5 global_prefetch_b8 for the next K panels
      __builtin_prefetch(&A[(size_t)(bm + ar) * K + kb + 16 + ac], 0, 1);
      __builtin_prefetch(&Bm[(size_t)(kb + 16 + br) * N + bn + bc], 0, 1);
    }
#if HAVE_ASYNC_LDS
    // Async DMA global->LDS (no VGPR round-trip, tracked by ASYNCcnt)
    __builtin_amdgcn_global_load_async_to_lds_b128(
        AS_GLOBAL(&A[(size_t)(bm + ar) * K + kb + ac]),
        AS_LDS(&sA[ar * 20 + ac]), 0, 0);
    __builtin_amdgcn_global_load_async_to_lds_b128(
        AS_GLOBAL(&Bm[(size_t)(kb + br) * N + bn + bc]),
        AS_LDS(&sB[br * 80 + bc]), 0, 0);
    __builtin_amdgcn_s_wait_asynccnt(0);
#else
    v4f av = *(const v4f*)&A[(size_t)(bm + ar) * K + kb + ac];
    v4f bv = *(const v4f*)&Bm[(size_t)(kb + br) * N + bn + bc];
    sA[ar * 20 + ac + 0] = av.x;
    sA[ar * 20 + ac + 1] = av.y;
    sA[ar * 20 + ac + 2] = av.z;
    sA[ar * 20 + ac + 3] = av.w;
    *(v4f*)&sB[br * 80 + bc] = bv;
#endif
    __syncthreads();

#pragma unroll
    for (int ks = 0; ks < 4; ++ks) {
      const int k0 = ks * 4 + 2 * lh;
      v2f a;
      a.x = sA[(wm + lm) * 20 + k0];
      a.y = sA[(wm + lm) * 20 + k0 + 1];
      v2f b0;
      b0.x = sB[k0 * 80 + wn + lm];
      b0.y = sB[(k0 + 1) * 80 + wn + lm];
      v2f b1;
      b1.x = sB[k0 * 80 + wn + 16 + lm];
      b1.y = sB[(k0 + 1) * 80 + wn + 16 + lm];
      acc0 = wmma4(a, b0, acc0);
      acc1 = wmma4(a, b1, acc1);
    }
    __syncthreads();
  }

#pragma unroll
  for (int r = 0; r < 8; ++r) {
    const int row = bm + wm + r + 8 * lh;
    const int c0  = bn + wn + lm;
    C[(size_t)row * N + c0]      = acc0[r] + bias[c0];
    C[(size_t)row * N + c0 + 16] = acc1[r] + bias[c0 + 16];
  }
}

// ---------------------------------------------------------------------------
// Attention for one (b, h, 16-query-row tile).
// 128 threads = 4 waves. Full score strip [16 x 2048] lives in dynamic LDS
// (128KB; gfx1250 WGP supports 320KB/workgroup). Exact softmax.
// ---------------------------------------------------------------------------
__global__ __launch_bounds__(128) void attn_kernel(
    const float* __restrict__ Q, const float* __restrict__ Kt,
    const float* __restrict__ V, float* __restrict__ O)
{
  extern __shared__ float smem[];
  float* sS     = smem;                 // 16 * 2048 scores
  float* sQ     = sS + 16 * SS;         // 16 * 72 padded Q tile
  float* red    = sQ + 16 * 72;         // 16 * 8 partial reductions
  float* rowmax = red + 128;            // 16
  float* rowinv = rowmax + 16;          // 16

  const int t0   = blockIdx.x * 16;
  const int h    = blockIdx.y;
  const int b    = blockIdx.z;
  const int tid  = threadIdx.x;
  const int lane = tid & 31;
  const int w    = tid >> 5;
  const int lm   = lane & 15;
  const int lh   = lane >> 4;

  const float* Qb = Q  + (size_t)b * SS * HSTR + h * HDM;
  const float* Kb = Kt + (size_t)b * SS * HSTR + h * HDM;
  const float* Vb = V  + (size_t)b * SS * HSTR + h * HDM;

  // ---- load Q tile [16 x 64] into LDS (stride 72 keeps frags aligned)
  {
    const int row = tid >> 3;
    const int c0  = (tid & 7) * 8;
    const float* src = &Qb[(size_t)(t0 + row) * HSTR + c0];
#if HAVE_ASYNC_LDS
    __builtin_amdgcn_global_load_async_to_lds_b128(
        AS_GLOBAL(src), AS_LDS(&sQ[row * 72 + c0]), 0, 0);
    __builtin_amdgcn_global_load_async_to_lds_b128(
        AS_GLOBAL(src + 4), AS_LDS(&sQ[row * 72 + c0 + 4]), 0, 0);
    __builtin_amdgcn_s_wait_asynccnt(0);
#else
#pragma unroll
    for (int i = 0; i < 8; ++i)
      sQ[row * 72 + c0 + i] = src[i];
#endif
  }
  __syncthreads();

  // ---- scores: S[16 x 2048] = Q (16x64) * K^T, scaled by 1/sqrt(64)
  for (int st = w; st < SS / 16; st += 4) {
    v8f acc = {};
#pragma unroll
    for (int ks = 0; ks < 16; ++ks) {
      const int kd = ks * 4 + 2 * lh;
      v2f a;
      a.x = sQ[lm * 72 + kd];
      a.y = sQ[lm * 72 + kd + 1];
      // B frag: B[k][n] = K[s_n][k]; lane n = lm selects key row
      v2f bf = *(const v2f*)&Kb[(size_t)(st * 16 + lm) * HSTR + kd];
      acc = wmma4(a, bf, acc);
    }
#pragma unroll
    for (int r = 0; r < 8; ++r) {
      const int m = r + 8 * lh;
      sS[m * SS + st * 16 + lm] = acc[r] * 0.125f;
    }
  }
  __syncthreads();

  // ---- softmax over s (exact, max-subtracted); 8 threads per row,
  //      float4 LDS accesses (rows are 8KB-strided, segments 16B-aligned)
  {
    const int row = tid >> 3;
    const int seg = tid & 7;
    v4f* p4 = (v4f*)(sS + row * SS + seg * 256);
    float mx = -3.402823466e38f;
    for (int i = 0; i < 64; ++i) {
      const v4f v = p4[i];
      mx = fmaxf(mx, fmaxf(fmaxf(v.x, v.y), fmaxf(v.z, v.w)));
    }
    red[row * 8 + seg] = mx;
    __syncthreads();
    if (tid < 16) {
      float m2 = red[tid * 8];
#pragma unroll
      for (int i = 1; i < 8; ++i) m2 = fmaxf(m2, red[tid * 8 + i]);
      rowmax[tid] = m2;
    }
    __syncthreads();
    const float m3 = rowmax[row];
    float sum = 0.f;
    for (int i = 0; i < 64; ++i) {
      v4f v = p4[i];
      v.x = expf(v.x - m3);
      v.y = expf(v.y - m3);
      v.z = expf(v.z - m3);
      v.w = expf(v.w - m3);
      p4[i] = v;
      sum += (v.x + v.y) + (v.z + v.w);
    }
    red[row * 8 + seg] = sum;
    __syncthreads();
    if (tid < 16) {
      float s2 = 0.f;
#pragma unroll
      for (int i = 0; i < 8; ++i) s2 += red[tid * 8 + i];
      rowinv[tid] = 1.0f / s2;
    }
    __syncthreads();
  }

  // ---- O tile [16 x 64] = P (16x2048) * V (2048x64); each wave owns 16 cols
  {
    const int n0 = w * 16;
    v8f acc = {};
    for (int kt = 0; kt < SS / 16; ++kt) {
#pragma unroll
      for (int ks = 0; ks < 4; ++ks) {
        const int s0 = kt * 16 + ks * 4 + 2 * lh;
        v2f a;
        a.x = sS[lm * SS + s0];
        a.y = sS[lm * SS + s0 + 1];
        v2f bf;
        bf.x = Vb[(size_t)s0 * HSTR + n0 + lm];
        bf.y = Vb[(size_t)(s0 + 1) * HSTR + n0 + lm];
        acc = wmma4(a, bf, acc);
      }
    }
#pragma unroll
    for (int r = 0; r < 8; ++r) {
      const int m = r + 8 * lh;
      O[((size_t)b * SS + t0 + m) * HSTR + h * HDM + n0 + lm] = acc[r] * rowinv[m];
    }
  }
}

// ---------------------------------------------------------------------------
// y = (1 - lambda) * o1 + lambda * o2, lambda derived on-device
// ---------------------------------------------------------------------------
__global__ void blend_kernel(const float* __restrict__ o1,
                             const float* __restrict__ o2,
                             const float* __restrict__ lambda_param,
                             const int* __restrict__ layer_idx,
                             float* __restrict__ y, int n)
{
  const int i = blockIdx.x * blockDim.x + threadIdx.x;
  if (i >= n) return;
  const float lf  = fminf(fmaxf((float)layer_idx[0] * 0.3f, 0.0f), 5.0f);
  const float off = 0.6f * expf(-lf);
  const float sig = 1.0f / (1.0f + expf(-lambda_param[0]));
  float lam = sig * (1.0f - off) + 0.2f;
  lam = fminf(fmaxf(lam, 0.1f), 0.9f);
  y[i] = (1.0f - lam) * o1[i] + lam * o2[i];
}

// ---------------------------------------------------------------------------
extern "C" void kernel_launch(void* const* d_in, const int* in_sizes, int n_in,
                              void* d_out, int out_size, void* d_ws, size_t ws_size,
                              hipStream_t stream)
{
  (void)in_sizes; (void)n_in; (void)out_size; (void)ws_size;
  // setup_inputs() dict order:
  const float* x            = (const float*)d_in[0];
  const float* proj_w       = (const float*)d_in[1];
  const float* proj_b       = (const float*)d_in[2];
  const float* lambda_param = (const float*)d_in[3];
  const int*   layer_idx    = (const int*)d_in[4];
  const float* wq1 = (const float*)d_in[5];
  const float* wk1 = (const float*)d_in[6];
  const float* wv1 = (const float*)d_in[7];
  const float* wo1 = (const float*)d_in[8];
  const float* bq1 = (const float*)d_in[9];
  const float* bk1 = (const float*)d_in[10];
  const float* bv1 = (const float*)d_in[11];
  const float* bo1 = (const float*)d_in[12];
  const float* wq2 = (const float*)d_in[13];
  const float* wk2 = (const float*)d_in[14];
  const float* wv2 = (const float*)d_in[15];
  const float* wo2 = (const float*)d_in[16];
  const float* bq2 = (const float*)d_in[17];
  const float* bk2 = (const float*)d_in[18];
  const float* bv2 = (const float*)d_in[19];
  const float* bo2 = (const float*)d_in[20];

  const size_t slot = (size_t)MROWS * DIMD;  // 4M floats per buffer
  float* ws = (float*)d_ws;
  float* q1 = ws + 0 * slot;
  float* k1 = ws + 1 * slot;
  float* v1 = ws + 2 * slot;
  float* q2 = ws + 3 * slot;
  float* k2 = ws + 4 * slot;
  float* v2 = ws + 5 * slot;
  float* o1 = ws + 6 * slot;   // attention outputs
  float* o2 = ws + 7 * slot;
  float* p1 = q1;              // reuse: per-branch projected outputs
  float* p2 = k1;
  float* yb = v1;              // blended

  const dim3 gg(DIMD / 64, MROWS / 64);   // (16, 64)

  // QKV projections (both branches)
  gemm_bias_kernel<<<gg, 256, 0, stream>>>(x, wq1, bq1, q1, MROWS, DIMD, DIMD);
  gemm_bias_kernel<<<gg, 256, 0, stream>>>(x, wk1, bk1, k1, MROWS, DIMD, DIMD);
  gemm_bias_kernel<<<gg, 256, 0, stream>>>(x, wv1, bv1, v1, MROWS, DIMD, DIMD);
  gemm_bias_kernel<<<gg, 256, 0, stream>>>(x, wq2, bq2, q2, MROWS, DIMD, DIMD);
  gemm_bias_kernel<<<gg, 256, 0, stream>>>(x, wk2, bk2, k2, MROWS, DIMD, DIMD);
  gemm_bias_kernel<<<gg, 256, 0, stream>>>(x, wv2, bv2, v2, MROWS, DIMD, DIMD);

  // Attention per (b, h, 16-row query tile)
  const dim3 ga(SS / 16, NH, BB);
  const size_t smem = (size_t)(16 * SS + 16 * 72 + 128 + 32) * sizeof(float);
  attn_kernel<<<ga, 128, smem, stream>>>(q1, k1, v1, o1);
  attn_kernel<<<ga, 128, smem, stream>>>(q2, k2, v2, o2);

  // Per-branch output projections (wo is [H*HD, DIM] row-major -> plain GEMM)
  gemm_bias_kernel<<<gg, 256, 0, stream>>>(o1, wo1, bo1, p1, MROWS, DIMD, DIMD);
  gemm_bias_kernel<<<gg, 256, 0, stream>>>(o2, wo2, bo2, p2, MROWS, DIMD, DIMD);

  // Blend with lambda
  const int n = MROWS * DIMD;
  blend_kernel<<<(n + 255) / 256, 256, 0, stream>>>(p1, p2, lambda_param, layer_idx, yb, n);

  // Final projection -> d_out
  gemm_bias_kernel<<<gg, 256, 0, stream>>>(yb, proj_w, proj_b, (float*)d_out, MROWS, DIMD, DIMD);
}